// Assoc_scan_DiagonalRNN_84232898609664
// MI455X (gfx1250) — compile-verified
//
#include <hip/hip_runtime.h>
#include <hip/hip_bf16.h>

// Diagonal linear RNN scan: h_t = a_p * h_{t-1} + DT * x_t, a_p = 1 - DT*relu(w_p)
// Shapes: x (B=32, L=8192, P=128) f32, w (P,) f32, out (B, L, P) f32.
//
// Roofline: 2 FLOP per element vs 12 B/element moved -> pure HBM-bandwidth
// problem. x is 134 MB and MI455X L2 is 192 MB, so the 3-phase scheme's second
// read of x in K3 should hit L2 (K1 just streamed it). Output stores are
// non-temporal so the out stream doesn't evict the L2-resident x. Effective
// HBM traffic ~= 134 MB read + 134 MB write ~= 11.5 us at 23.3 TB/s.
//
// WMMA is inapplicable: the 16x16 scan matrix L(a_p) differs per channel,
// while one WMMA applies a single matrix per wave. The CDNA5-specific path
// that fits is async global->LDS staging (ASYNCcnt) with double buffering.

#define DT_CONST 1.0f

constexpr int B_   = 32;
constexpr int L_   = 8192;
constexpr int P_   = 128;
constexpr int NCH  = 128;          // chunks per (b,p) sequence
constexpr int CLEN = L_ / NCH;     // 64 timesteps per chunk
constexpr int SUB  = 16;           // timesteps per LDS-staged subtile
constexpr int NSUB = CLEN / SUB;   // 4 subtiles per chunk
constexpr int TILE_FLOATS = SUB * P_;   // 2048 floats = 8 KB per subtile buffer

// ---- CDNA5 async-to-LDS availability (guarded: fallback keeps compile clean) ----
#if defined(__HIP_DEVICE_COMPILE__) && defined(__gfx1250__) && \
    __has_builtin(__builtin_amdgcn_global_load_async_to_lds_b128) && \
    __has_builtin(__builtin_amdgcn_s_wait_asynccnt)
#define USE_ASYNC_LDS 1
#else
#define USE_ASYNC_LDS 0
#endif

#if USE_ASYNC_LDS
// Pointee type taken from the builtin's signature (clang diagnostic):
// 'int __attribute__((vector_size(16)))'
typedef int v4i_ __attribute__((vector_size(16)));
typedef __attribute__((address_space(1))) v4i_* g_v4i_ptr;
typedef __attribute__((address_space(3))) v4i_* l_v4i_ptr;
#endif

__device__ __forceinline__ float decay_coef(const float* w, int p) {
    return 1.0f - DT_CONST * fmaxf(w[p], 0.0f);
}

// ---------------- Kernel 1: per-chunk local reduction (zero initial state) ----------------
// grid = B*NCH blocks, block = P threads (thread p owns channel p).
// Default (RT) loads: deliberately leave x resident in the 192 MB L2 for K3.
__global__ void k1_chunk_summary(const float* __restrict__ x,
                                 const float* __restrict__ w,
                                 float* __restrict__ ws) {
    const int p   = threadIdx.x;
    const int bid = blockIdx.x;
    const int b   = bid / NCH;
    const int c   = bid % NCH;
    const float a = decay_coef(w, p);

    const float* xbase = x + ((size_t)b * L_ + (size_t)c * CLEN) * P_;
    float s = 0.0f;
    #pragma unroll 8
    for (int t = 0; t < CLEN; ++t) {
        s = fmaf(a, s, DT_CONST * xbase[(size_t)t * P_ + p]);   // coalesced 512B/step
    }
    ws[((size_t)b * NCH + c) * P_ + p] = s;
}

// ---------------- Kernel 2: scan chunk summaries -> incoming carries ----------------
// B*P = 4096 threads total; thread (b,p) serially scans its NCH summaries.
__global__ void k2_carry_scan(const float* __restrict__ w,
                              float* __restrict__ ws) {
    const int tid = blockIdx.x * blockDim.x + threadIdx.x;
    const int b = tid / P_;
    const int p = tid % P_;
    const float a = decay_coef(w, p);

    float aC = a;                       // a^(2^6) = a^64 = a^CLEN
    #pragma unroll
    for (int i = 0; i < 6; ++i) aC *= aC;

    float* base = ws + (size_t)b * NCH * P_ + p;
    float carry = 0.0f;
    for (int c = 0; c < NCH; ++c) {
        const float sc = base[(size_t)c * P_];  // coalesced across p
        base[(size_t)c * P_] = carry;           // incoming state for chunk c
        carry = fmaf(aC, carry, sc);            // state at end of chunk c
    }
}

// ---------------- Kernel 3: apply carries, emit full sequence ----------------
// grid = B*NCH blocks, block = P threads. x staged through LDS with CDNA5
// async-to-LDS double buffering; each wave tracks its own ASYNCcnt (in-order
// completion), barriers make cross-wave LDS writes visible. Output stores are
// non-temporal to preserve x's L2 residency.
__global__ void k3_apply(const float* __restrict__ x,
                         const float* __restrict__ w,
                         const float* __restrict__ ws,
                         float* __restrict__ out) {
    const int p   = threadIdx.x;
    const int bid = blockIdx.x;
    const int b   = bid / NCH;
    const int c   = bid % NCH;
    const float a = decay_coef(w, p);

    float h = ws[((size_t)b * NCH + c) * P_ + p];   // incoming state for this chunk

    const float* xbase = x   + ((size_t)b * L_ + (size_t)c * CLEN) * P_;
    float*       obase = out + ((size_t)b * L_ + (size_t)c * CLEN) * P_;

#if USE_ASYNC_LDS
    __shared__ float lds[2][TILE_FLOATS];

    auto issue = [&](int s) {
        const float* gsrc = xbase + (size_t)s * TILE_FLOATS;
        float*       ldst = &lds[s & 1][0];
        #pragma unroll
        for (int i = 0; i < TILE_FLOATS / (P_ * 4); ++i) {      // 4 issues x 16B/thread
            const int off = (i * P_ + p) * 4;                   // float index
            __builtin_amdgcn_global_load_async_to_lds_b128(
                (g_v4i_ptr)(gsrc + off),
                (l_v4i_ptr)(ldst + off),
                /*offset=*/0, /*cpol=*/0);
        }
    };

    issue(0);
    #pragma unroll
    for (int s = 0; s < NSUB; ++s) {
        if (s + 1 < NSUB) {
            issue(s + 1);                               // prefetch next subtile
            __builtin_amdgcn_s_wait_asynccnt(4);        // subtile s done (in-order), next 4 in flight
        } else {
            __builtin_amdgcn_s_wait_asynccnt(0);
        }
        __syncthreads();                                // all waves' async writes visible

        const float* buf = &lds[s & 1][0];
        float*       ob  = obase + (size_t)s * TILE_FLOATS;
        #pragma unroll
        for (int t = 0; t < SUB; ++t) {
            const float xv = buf[t * P_ + p];           // bank-conflict-free
            h = fmaf(a, h, DT_CONST * xv);
            __builtin_nontemporal_store(h, &ob[(size_t)t * P_ + p]);  // NT: don't evict x from L2
        }
        __syncthreads();                                // buffer reusable for s+2
    }
#else
    for (int t = 0; t < CLEN; ++t) {
        h = fmaf(a, h, DT_CONST * xbase[(size_t)t * P_ + p]);
        __builtin_nontemporal_store(h, &obase[(size_t)t * P_ + p]);
    }
#endif
}

extern "C" void kernel_launch(void* const* d_in, const int* in_sizes, int n_in,
                              void* d_out, int out_size, void* d_ws, size_t ws_size,
                              hipStream_t stream) {
    const float* x = (const float*)d_in[0];   // (B, L, P) f32
    const float* w = (const float*)d_in[1];   // (P,) f32
    float* out = (float*)d_out;               // (B, L, P) f32
    float* ws  = (float*)d_ws;                // needs B*NCH*P*4 = 2 MB

    (void)in_sizes; (void)n_in; (void)out_size; (void)ws_size;

    k1_chunk_summary<<<dim3(B_ * NCH), dim3(P_), 0, stream>>>(x, w, ws);
    k2_carry_scan<<<dim3((B_ * P_) / 128), dim3(128), 0, stream>>>(w, ws);
    k3_apply<<<dim3(B_ * NCH), dim3(P_), 0, stream>>>(x, w, ws, out);
}